// NonStationaryTransformer_57114475102268
// MI455X (gfx1250) — compile-verified
//
#include <hip/hip_runtime.h>
#include <math.h>

#define B_    32
#define SEQ_  512
#define LBL_  96
#define PRD_  96
#define EIN_  64
#define D_    128
#define H_    4
#define E_    32
#define DFF_  256
#define HID_  64
#define DLEN_ 192
#define EPS_  1e-5f

typedef __attribute__((ext_vector_type(2))) float v2f;
typedef __attribute__((ext_vector_type(8))) float v8f;

// Exact fp32 WMMA: D = A(16x4) * B(4x16) + C(16x16), all f32.
static __device__ inline v8f wmma_k4(v2f a, v2f b, v8f c) {
  return __builtin_amdgcn_wmma_f32_16x16x4_f32(false, a, false, b, (short)0, c,
                                               false, false);
}

// ---------------------------------------------------------------------------
// LDS-staged, double-buffered GEMM: C[M,N] = act(A[M,K] @ W[K,N] + bias).
// Compile-time N,K -> immediate addressing, fully unrolled chunk loop.
// Workgroup = 4 waves, 64x64 tile; KC=32 K-chunks, ping-pong LDS buffers:
//   As: row-major, stride 36 (16B-aligned float4 staging, conflict-free frags)
//   Bs: transposed (Bs[n*34 + k]) so the (k,k+1) fragment is one ds_load_b64.
// Pipeline per chunk: issue next chunk's global loads -> WMMA on current
// buffer (hides global latency) -> LDS stores to other buffer -> one barrier.
// __launch_bounds__(128,1): let RA use the full VGPR budget (no spills);
// latency is hidden by the LDS pipeline, not occupancy.
// act: 0 = none, 1 = exact GELU.
// ---------------------------------------------------------------------------
template<int N, int K>
__global__ __launch_bounds__(128, 1) void k_gemm_t(
    const float* __restrict__ A, const float* __restrict__ W,
    const float* __restrict__ bias, float* __restrict__ C,
    int M, int act)
{
  constexpr int KC  = 32;
  constexpr int STA = 36;
  constexpr int STB = 34;
  constexpr int NC  = K / KC;
  __shared__ float As[2][64 * STA];
  __shared__ float Bs[2][64 * STB];

  const int tid  = threadIdx.x;
  const int lane = tid & 31;
  const int wave = tid >> 5;
  const int ni   = lane & 15;   // row-in-tile for A, col-in-tile for B/C
  const int hl   = lane >> 4;   // half-wave select
  const int m0 = blockIdx.y * 64;
  const int n0 = blockIdx.x * 64;

  // Per-thread staging slots (fixed across chunks).
  const int arow = tid >> 1;                 // A: 2 threads per row
  const int ac4  = (tid & 1) << 4;           // each covers 16 cols (4 float4)
  const int bkr  = tid >> 2;                 // B: 4 threads per k-row
  const int bn4  = (tid & 3) << 4;           // each covers 16 cols

  float4 ra[4], rb[4];
  auto load_regs = [&](int kc) {
    const float* Ap = A + (size_t)(m0 + arow) * K + kc + ac4;
    #pragma unroll
    for (int i = 0; i < 4; ++i) ra[i] = *(const float4*)(Ap + i * 4);
    const float* Wp = W + (size_t)(kc + bkr) * N + n0 + bn4;
    #pragma unroll
    for (int i = 0; i < 4; ++i) rb[i] = *(const float4*)(Wp + i * 4);
  };
  auto store_lds = [&](int buf) {
    float* as = &As[buf][arow * STA + ac4];
    #pragma unroll
    for (int i = 0; i < 4; ++i) *(float4*)(as + i * 4) = ra[i];
    float* bs = &Bs[buf][bn4 * STB + bkr];
    #pragma unroll
    for (int i = 0; i < 4; ++i) {
      bs[(i * 4 + 0) * STB] = rb[i].x;
      bs[(i * 4 + 1) * STB] = rb[i].y;
      bs[(i * 4 + 2) * STB] = rb[i].z;
      bs[(i * 4 + 3) * STB] = rb[i].w;
    }
  };

  v8f acc[4];
  #pragma unroll
  for (int t = 0; t < 4; ++t) acc[t] = (v8f){0.f,0.f,0.f,0.f,0.f,0.f,0.f,0.f};

  load_regs(0);
  store_lds(0);
  __syncthreads();

  #pragma unroll
  for (int c = 0; c < NC; ++c) {
    if (c + 1 < NC) load_regs((c + 1) * KC);   // in flight during compute
    const float* Ab = &As[c & 1][(wave * 16 + ni) * STA + hl * 2];
    const float* Bb = &Bs[c & 1][ni * STB + hl * 2];
    #pragma unroll
    for (int k = 0; k < KC; k += 4) {
      float2 af = *(const float2*)(Ab + k);
      v2f a; a.x = af.x; a.y = af.y;
      #pragma unroll
      for (int t = 0; t < 4; ++t) {
        float2 bf = *(const float2*)(Bb + t * 16 * STB + k);
        v2f b; b.x = bf.x; b.y = bf.y;
        acc[t] = wmma_k4(a, b, acc[t]);
      }
    }
    if (c + 1 < NC) store_lds((c + 1) & 1);    // other buffer: race-free
    __syncthreads();
  }

  const int mw = m0 + wave * 16;
  #pragma unroll
  for (int t = 0; t < 4; ++t) {
    const int col = n0 + t * 16 + ni;
    const float bv = bias ? bias[col] : 0.0f;
    #pragma unroll
    for (int r = 0; r < 8; ++r) {
      const int row = mw + hl * 8 + r;
      float v = acc[t][r] + bv;
      if (act == 1) v = 0.5f * v * (1.0f + erff(v * 0.70710678118654752f));
      C[(size_t)row * N + col] = v;
    }
  }
}

// ---------------------------------------------------------------------------
// Fused attention: one wave per (b, h, 16-row q tile).
// scores = (Q Kt) * tau[b] (+ delta[b,s]) / sqrt(E); softmax over S; out = A V.
// Q/K/V/O are [B*L, D] with head h at columns [h*32, h*32+32).
// Score rows live in LDS (16 x S, stride 516 to avoid bank conflicts).
// ---------------------------------------------------------------------------
template<int S>
__global__ __launch_bounds__(32) void k_attn_t(
    const float* __restrict__ Q, const float* __restrict__ K,
    const float* __restrict__ V, float* __restrict__ O,
    const float* __restrict__ tau, const float* __restrict__ delta,
    int Lq)
{
  constexpr int SP = 516;
  const int lane = threadIdx.x & 31;
  const int ni   = lane & 15;
  const int hl   = lane >> 4;
  const int l0   = blockIdx.x * 16;
  const int h    = blockIdx.y;
  const int b    = blockIdx.z;
  const float tauv = tau[b];
  const float invs = 0.17677669529663687f;  // 1/sqrt(32)

  __shared__ float sc[16 * SP];

  // Preload the wave's 16 q rows as 8 WMMA fragments (reused for all S tiles).
  const float* qrow = Q + ((size_t)b * Lq + l0 + ni) * D_ + h * E_ + hl * 2;
  v2f qa[8];
  #pragma unroll
  for (int i = 0; i < 8; ++i) {
    float2 t = *(const float2*)(qrow + i * 4);
    qa[i].x = t.x; qa[i].y = t.y;
  }

  // Phase 1: score tiles via WMMA (A = q rows, B = K^T columns).
  const size_t kbase = (size_t)b * S * D_ + h * E_ + hl * 2;
  #pragma unroll 2
  for (int st = 0; st < S; st += 16) {
    v8f acc = (v8f){0.f,0.f,0.f,0.f,0.f,0.f,0.f,0.f};
    const float* krow = K + kbase + (size_t)(st + ni) * D_;
    #pragma unroll
    for (int k = 0; k < E_; k += 4) {
      float2 bk = *(const float2*)(krow + k);
      v2f bb; bb.x = bk.x; bb.y = bk.y;
      acc = wmma_k4(qa[k >> 2], bb, acc);
    }
    const int s = st + ni;
    const float dv = delta ? delta[(size_t)b * SEQ_ + s] : 0.f;
    #pragma unroll
    for (int r = 0; r < 8; ++r)
      sc[(hl * 8 + r) * SP + s] = (acc[r] * tauv + dv) * invs;
  }
  __syncthreads();

  // Phase 2: row softmax; 2 lanes per row (lane, lane^16), xor-combine.
  {
    float* row = sc + ni * SP;
    float mx = -3.402823466e38f;
    for (int s = hl; s < S; s += 2) mx = fmaxf(mx, row[s]);
    mx = fmaxf(mx, __shfl_xor(mx, 16, 32));
    float sum = 0.f;
    for (int s = hl; s < S; s += 2) { float e = expf(row[s] - mx); row[s] = e; sum += e; }
    sum += __shfl_xor(sum, 16, 32);
    const float inv = 1.f / sum;
    for (int s = hl; s < S; s += 2) row[s] *= inv;
  }
  __syncthreads();

  // Phase 3: out = A @ V via WMMA (A fragments straight from LDS).
  #pragma unroll
  for (int nt = 0; nt < 2; ++nt) {
    v8f acc = (v8f){0.f,0.f,0.f,0.f,0.f,0.f,0.f,0.f};
    const int col = h * E_ + nt * 16 + ni;
    const float* sb = sc + ni * SP + hl * 2;
    #pragma unroll 4
    for (int k = 0; k < S; k += 4) {
      float2 as = *(const float2*)(sb + k);
      v2f a; a.x = as.x; a.y = as.y;
      const float* vp = V + ((size_t)b * S + k + hl * 2) * D_ + col;
      v2f bb; bb.x = vp[0]; bb.y = vp[D_];
      acc = wmma_k4(a, bb, acc);
    }
    #pragma unroll
    for (int r = 0; r < 8; ++r)
      O[((size_t)b * Lq + l0 + hl * 8 + r) * D_ + col] = acc[r];
  }
}

// ---------------------------------------------------------------------------
// LayerNorm over D=128: Out = LN(X + Y) * g + b   (Y may be null -> LN(X)).
// One wave per row; float4 per lane; xor-shuffle reductions (wave32).
// ---------------------------------------------------------------------------
__global__ __launch_bounds__(128) void k_ln(
    const float* __restrict__ X, const float* __restrict__ Y,
    const float* __restrict__ g, const float* __restrict__ bta,
    float* __restrict__ O, int M)
{
  const int row = blockIdx.x * 4 + (threadIdx.x >> 5);
  if (row >= M) return;
  const int lane = threadIdx.x & 31;
  const int c = lane * 4;
  float4 v = *(const float4*)(X + (size_t)row * D_ + c);
  if (Y) {
    float4 y = *(const float4*)(Y + (size_t)row * D_ + c);
    v.x += y.x; v.y += y.y; v.z += y.z; v.w += y.w;
  }
  float s = v.x + v.y + v.z + v.w;
  #pragma unroll
  for (int m = 16; m >= 1; m >>= 1) s += __shfl_xor(s, m, 32);
  const float mean = s * (1.f / D_);
  const float d0 = v.x - mean, d1 = v.y - mean, d2 = v.z - mean, d3 = v.w - mean;
  float q = d0 * d0 + d1 * d1 + d2 * d2 + d3 * d3;
  #pragma unroll
  for (int m = 16; m >= 1; m >>= 1) q += __shfl_xor(q, m, 32);
  const float rs = rsqrtf(q * (1.f / D_) + EPS_);
  float4 o;
  o.x = d0 * rs * g[c + 0] + bta[c + 0];
  o.y = d1 * rs * g[c + 1] + bta[c + 1];
  o.z = d2 * rs * g[c + 2] + bta[c + 2];
  o.w = d3 * rs * g[c + 3] + bta[c + 3];
  *(float4*)(O + (size_t)row * D_ + c) = o;
}

// --------------------------- small VALU kernels ----------------------------

__global__ void k_stats(const float* __restrict__ x, float* __restrict__ mu,
                        float* __restrict__ sig)
{
  const int idx = blockIdx.x * blockDim.x + threadIdx.x;
  if (idx >= B_ * EIN_) return;
  const int b = idx / EIN_, c = idx % EIN_;
  const float* p = x + (size_t)b * SEQ_ * EIN_ + c;
  float s = 0.f;
  for (int t = 0; t < SEQ_; ++t) s += p[(size_t)t * EIN_];
  const float m = s * (1.f / SEQ_);
  float q = 0.f;
  for (int t = 0; t < SEQ_; ++t) { float d = p[(size_t)t * EIN_] - m; q += d * d; }
  mu[idx]  = m;
  sig[idx] = sqrtf(q * (1.f / SEQ_) + EPS_);
}

__global__ void k_norm_in(const float* __restrict__ x, const float* __restrict__ mu,
                          const float* __restrict__ sig, float* __restrict__ xe)
{
  const int idx = blockIdx.x * blockDim.x + threadIdx.x;
  if (idx >= B_ * SEQ_ * EIN_) return;
  const int b = idx / (SEQ_ * EIN_), c = idx % EIN_;
  xe[idx] = (x[idx] - mu[b * EIN_ + c]) / sig[b * EIN_ + c];
}

__global__ void k_dec_in(const float* __restrict__ xe, float* __restrict__ di)
{
  const int idx = blockIdx.x * blockDim.x + threadIdx.x;
  if (idx >= B_ * DLEN_ * EIN_) return;
  const int e = idx % EIN_;
  const int l = (idx / EIN_) % DLEN_;
  const int b = idx / (EIN_ * DLEN_);
  di[idx] = (l < LBL_) ? xe[((size_t)b * SEQ_ + (SEQ_ - LBL_) + l) * EIN_ + e] : 0.f;
}

// Circular Conv1d(SEQ channels -> 1, kernel 3) over length EIN.
__global__ void k_conv(const float* __restrict__ x, const float* __restrict__ w,
                       float* __restrict__ out)
{
  const int idx = blockIdx.x * blockDim.x + threadIdx.x;
  if (idx >= B_ * EIN_) return;
  const int b = idx / EIN_, e = idx % EIN_;
  const int em1 = (e + EIN_ - 1) % EIN_, ep1 = (e + 1) % EIN_;
  float acc = 0.f;
  for (int c = 0; c < SEQ_; ++c) {
    const float* xr = x + ((size_t)b * SEQ_ + c) * EIN_;
    acc += xr[em1] * w[c * 3 + 0] + xr[e] * w[c * 3 + 1] + xr[ep1] * w[c * 3 + 2];
  }
  out[idx] = acc;
}

__global__ void k_mlp_hid(const float* __restrict__ conv, const float* __restrict__ stat,
                          const float* __restrict__ w0, const float* __restrict__ b0,
                          float* __restrict__ hid)
{
  const int idx = blockIdx.x * blockDim.x + threadIdx.x;
  if (idx >= B_ * HID_) return;
  const int b = idx / HID_, hcol = idx % HID_;
  float acc = b0[hcol];
  for (int i = 0; i < EIN_; ++i) acc += conv[b * EIN_ + i] * w0[i * HID_ + hcol];
  for (int i = 0; i < EIN_; ++i) acc += stat[b * EIN_ + i] * w0[(EIN_ + i) * HID_ + hcol];
  hid[idx] = fmaxf(acc, 0.f);
}

__global__ void k_tau(const float* __restrict__ hid, const float* __restrict__ wo,
                      float* __restrict__ tau)
{
  const int b = blockIdx.x * blockDim.x + threadIdx.x;
  if (b >= B_) return;
  float acc = 0.f;
  for (int h = 0; h < HID_; ++h) acc += hid[b * HID_ + h] * wo[h];
  tau[b] = expf(acc);
}

__global__ void k_delta(const float* __restrict__ hid, const float* __restrict__ wo,
                        float* __restrict__ delta)
{
  const int idx = blockIdx.x * blockDim.x + threadIdx.x;
  if (idx >= B_ * SEQ_) return;
  const int b = idx / SEQ_, s = idx % SEQ_;
  float acc = 0.f;
  for (int h = 0; h < HID_; ++h) acc += hid[b * HID_ + h] * wo[h * SEQ_ + s];
  delta[idx] = acc;
}

// X += vb + tb + PE(l,d) + mark0*tw[0,d] + mark1*tw[1,d]
__global__ void k_embed_add(float* __restrict__ X, const float* __restrict__ vb,
                            const float* __restrict__ tw, const float* __restrict__ tb,
                            const float* __restrict__ mark, int L, int total)
{
  const int idx = blockIdx.x * blockDim.x + threadIdx.x;
  if (idx >= total) return;
  const int d = idx & (D_ - 1);
  const int row = idx >> 7;
  const int l = row % L;
  const float m0 = mark[(size_t)row * 2 + 0];
  const float m1 = mark[(size_t)row * 2 + 1];
  const int i2 = d & ~1;
  const float freq = expf(-(float)i2 * (9.210340371976184f / 128.0f));
  const float ang = (float)l * freq;
  const float pe = (d & 1) ? cosf(ang) : sinf(ang);
  X[idx] += vb[d] + tb[d] + pe + m0 * tw[d] + m1 * tw[D_ + d];
}

__global__ void k_proj(const float* __restrict__ dec, const float* __restrict__ pw,
                       const float* __restrict__ pb, float* __restrict__ ol)
{
  const int idx = blockIdx.x * blockDim.x + threadIdx.x;
  if (idx >= B_ * DLEN_) return;
  float acc = pb[0];
  const float* dp = dec + (size_t)idx * D_;
  for (int d = 0; d < D_; ++d) acc += dp[d] * pw[d];
  ol[idx] = acc;
}

__global__ void k_final(const float* __restrict__ ol, const float* __restrict__ sig,
                        const float* __restrict__ mu, float* __restrict__ out)
{
  const int idx = blockIdx.x * blockDim.x + threadIdx.x;
  if (idx >= B_ * PRD_ * EIN_) return;
  const int e = idx % EIN_;
  const int t = (idx / EIN_) % PRD_;
  const int b = idx / (EIN_ * PRD_);
  out[idx] = ol[b * DLEN_ + LBL_ + t] * sig[b * EIN_ + e] + mu[b * EIN_ + e];
}

// ------------------------- parameter index mapping -------------------------
// setup_inputs() dict flattened in insertion order (recursively).
enum {
  IN_XENC = 0, IN_MARK_ENC, IN_XDEC, IN_MARK_DEC,
  P_ENC_EMB_VW, P_ENC_EMB_VB, P_ENC_EMB_TW, P_ENC_EMB_TB,   // 4..7
  P_DEC_EMB_VW, P_DEC_EMB_VB, P_DEC_EMB_TW, P_DEC_EMB_TB,   // 8..11
  P_TAU_CONV, P_TAU_W0, P_TAU_B0, P_TAU_WO,                 // 12..15
  P_DELTA_CONV, P_DELTA_W0, P_DELTA_B0, P_DELTA_WO,         // 16..19
  P_ENC0 = 20,            // enc block: qw qb kw kb vw vb ow ob w1 b1 w2 b2 n1g n1b n2g n2b
  P_DEC0 = P_ENC0 + 32,   // 52; dec block adds: ca(qw..ob)=+16..+23, n3g=+24, n3b=+25
  P_ENC_NORM_G = P_DEC0 + 26,  // 78
  P_ENC_NORM_B, P_DEC_NORM_G, P_DEC_NORM_B, P_PROJ_W, P_PROJ_B  // 79..83
};

extern "C" void kernel_launch(void* const* d_in, const int* in_sizes, int n_in,
                              void* d_out, int out_size, void* d_ws, size_t ws_size,
                              hipStream_t stream)
{
  (void)in_sizes; (void)out_size; (void)ws_size;
  const float* P[96];
  for (int i = 0; i < n_in && i < 96; ++i) P[i] = (const float*)d_in[i];

  float* ws = (float*)d_ws;
  size_t off = 0;
  auto alloc = [&](size_t n) { float* p = ws + off; off += n; return p; };
  float* XE    = alloc((size_t)B_ * SEQ_ * EIN_);
  float* DIN   = alloc((size_t)B_ * DLEN_ * EIN_);
  float* EX    = alloc((size_t)B_ * SEQ_ * D_);
  float* DX    = alloc((size_t)B_ * DLEN_ * D_);
  float* Qb    = alloc((size_t)B_ * SEQ_ * D_);
  float* Kb    = alloc((size_t)B_ * SEQ_ * D_);
  float* Vb    = alloc((size_t)B_ * SEQ_ * D_);
  float* AO    = alloc((size_t)B_ * SEQ_ * D_);
  float* T0    = alloc((size_t)B_ * SEQ_ * D_);
  float* FH    = alloc((size_t)B_ * SEQ_ * DFF_);
  float* MU    = alloc(B_ * EIN_);
  float* SIG   = alloc(B_ * EIN_);
  float* TAU   = alloc(B_);
  float* DELTA = alloc((size_t)B_ * SEQ_);
  float* CONV  = alloc(B_ * EIN_);
  float* HIDB  = alloc(B_ * HID_);
  float* OL    = alloc((size_t)B_ * DLEN_);

  auto gemm = [&](const float* A, const float* W, const float* bias, float* C,
                  int M, int N, int K, int act) {
    dim3 g(N / 64, M / 64);
    if (N == 128 && K == 64)
      k_gemm_t<128, 64><<<g, 128, 0, stream>>>(A, W, bias, C, M, act);
    else if (N == 128 && K == 128)
      k_gemm_t<128, 128><<<g, 128, 0, stream>>>(A, W, bias, C, M, act);
    else if (N == 256 && K == 128)
      k_gemm_t<256, 128><<<g, 128, 0, stream>>>(A, W, bias, C, M, act);
    else if (N == 128 && K == 256)
      k_gemm_t<128, 256><<<g, 128, 0, stream>>>(A, W, bias, C, M, act);
  };
  auto attn = [&](const float* Q, const float* K, const float* V, float* O,
                  const float* dl, int Lq, int S) {
    dim3 g(Lq / 16, H_, B_);
    if (S == SEQ_) k_attn_t<SEQ_><<<g, 32, 0, stream>>>(Q, K, V, O, TAU, dl, Lq);
    else           k_attn_t<DLEN_><<<g, 32, 0, stream>>>(Q, K, V, O, TAU, dl, Lq);
  };
  auto ln = [&](const float* X, const float* Y, const float* g, const float* b,
                float* O, int M) {
    k_ln<<<dim3(M / 4), 128, 0, stream>>>(X, Y, g, b, O, M);
  };

  // Stats + normalization
  k_stats<<<dim3((B_ * EIN_ + 127) / 128), 128, 0, stream>>>(P[IN_XENC], MU, SIG);
  k_norm_in<<<dim3((B_ * SEQ_ * EIN_ + 255) / 256), 256, 0, stream>>>(P[IN_XENC], MU, SIG, XE);

  // tau projector (stats = sigma)
  k_conv<<<dim3((B_ * EIN_ + 127) / 128), 128, 0, stream>>>(P[IN_XENC], P[P_TAU_CONV], CONV);
  k_mlp_hid<<<dim3((B_ * HID_ + 127) / 128), 128, 0, stream>>>(CONV, SIG, P[P_TAU_W0], P[P_TAU_B0], HIDB);
  k_tau<<<dim3(1), 32, 0, stream>>>(HIDB, P[P_TAU_WO], TAU);
  // delta projector (stats = mu)
  k_conv<<<dim3((B_ * EIN_ + 127) / 128), 128, 0, stream>>>(P[IN_XENC], P[P_DELTA_CONV], CONV);
  k_mlp_hid<<<dim3((B_ * HID_ + 127) / 128), 128, 0, stream>>>(CONV, MU, P[P_DELTA_W0], P[P_DELTA_B0], HIDB);
  k_delta<<<dim3((B_ * SEQ_ + 255) / 256), 256, 0, stream>>>(HIDB, P[P_DELTA_WO], DELTA);

  // Decoder input (label slice + zero prediction window)
  k_dec_in<<<dim3((B_ * DLEN_ * EIN_ + 255) / 256), 256, 0, stream>>>(XE, DIN);

  // Embeddings
  const int Me = B_ * SEQ_, Md = B_ * DLEN_;
  gemm(XE, P[P_ENC_EMB_VW], nullptr, EX, Me, D_, EIN_, 0);
  k_embed_add<<<dim3((Me * D_ + 255) / 256), 256, 0, stream>>>(
      EX, P[P_ENC_EMB_VB], P[P_ENC_EMB_TW], P[P_ENC_EMB_TB], P[IN_MARK_ENC], SEQ_, Me * D_);
  gemm(DIN, P[P_DEC_EMB_VW], nullptr, DX, Md, D_, EIN_, 0);
  k_embed_add<<<dim3((Md * D_ + 255) / 256), 256, 0, stream>>>(
      DX, P[P_DEC_EMB_VB], P[P_DEC_EMB_TW], P[P_DEC_EMB_TB], P[IN_MARK_DEC], DLEN_, Md * D_);

  // Encoder blocks
  for (int blk = 0; blk < 2; ++blk) {
    const int p = P_ENC0 + blk * 16;
    gemm(EX, P[p + 0], P[p + 1], Qb, Me, D_, D_, 0);
    gemm(EX, P[p + 2], P[p + 3], Kb, Me, D_, D_, 0);
    gemm(EX, P[p + 4], P[p + 5], Vb, Me, D_, D_, 0);
    attn(Qb, Kb, Vb, AO, nullptr, SEQ_, SEQ_);
    gemm(AO, P[p + 6], P[p + 7], T0, Me, D_, D_, 0);
    ln(EX, T0, P[p + 12], P[p + 13], EX, Me);                 // n1
    gemm(EX, P[p + 8], P[p + 9], FH, Me, DFF_, D_, 1);        // GELU
    gemm(FH, P[p + 10], P[p + 11], T0, Me, D_, DFF_, 0);
    ln(EX, T0, P[p + 14], P[p + 15], EX, Me);                 // n2
  }
  ln(EX, nullptr, P[P_ENC_NORM_G], P[P_ENC_NORM_B], EX, Me);

  // Decoder block
  {
    const int p = P_DEC0;
    // self-attention
    gemm(DX, P[p + 0], P[p + 1], Qb, Md, D_, D_, 0);
    gemm(DX, P[p + 2], P[p + 3], Kb, Md, D_, D_, 0);
    gemm(DX, P[p + 4], P[p + 5], Vb, Md, D_, D_, 0);
    attn(Qb, Kb, Vb, AO, nullptr, DLEN_, DLEN_);
    gemm(AO, P[p + 6], P[p + 7], T0, Md, D_, D_, 0);
    ln(DX, T0, P[p + 12], P[p + 13], DX, Md);                 // n1
    // cross-attention (with delta bias on keys)
    gemm(DX, P[p + 16], P[p + 17], Qb, Md, D_, D_, 0);
    gemm(EX, P[p + 18], P[p + 19], Kb, Me, D_, D_, 0);
    gemm(EX, P[p + 20], P[p + 21], Vb, Me, D_, D_, 0);
    attn(Qb, Kb, Vb, AO, DELTA, DLEN_, SEQ_);
    gemm(AO, P[p + 22], P[p + 23], T0, Md, D_, D_, 0);
    ln(DX, T0, P[p + 14], P[p + 15], DX, Md);                 // n2
    // FFN
    gemm(DX, P[p + 8], P[p + 9], FH, Md, DFF_, D_, 1);        // GELU
    gemm(FH, P[p + 10], P[p + 11], T0, Md, D_, DFF_, 0);
    ln(DX, T0, P[p + 24], P[p + 25], DX, Md);                 // n3
  }
  ln(DX, nullptr, P[P_DEC_NORM_G], P[P_DEC_NORM_B], DX, Md);

  // Final projection + de-normalization broadcast
  k_proj<<<dim3((Md + 127) / 128), 128, 0, stream>>>(DX, P[P_PROJ_W], P[P_PROJ_B], OL);
  k_final<<<dim3((B_ * PRD_ * EIN_ + 255) / 256), 256, 0, stream>>>(
      OL, SIG, MU, (float*)d_out);
}